// Fold_2757369004157
// MI455X (gfx1250) — compile-verified
//
#include <hip/hip_runtime.h>

// Fold / transposed-conv3d with identity weights == 27-tap shift-and-add.
// x: (2, 27, 96, 96, 96) f32  ->  out: (2, 1, 96, 96, 96) f32
// Pure streaming: ~198 MB traffic -> ~8.5 us floor at 23.3 TB/s; compute is
// negligible, so the CDNA5 feature that matters is the data-mover path:
//   interior blocks: 3x tensor_load_to_lds (TDM 96x3x3 tiles, TENSORcnt)
//   boundary blocks: per-lane global_load_async_to_lds_b32 (ASYNCcnt)
// x-shifts resolved via branch-free clamped LDS reads; fp32-exact VALU adds.

#define DIM  96
#define NCH  27
#define CS   (DIM * DIM * DIM)        // channel stride, elements
#define NBLK (2 * DIM * DIM)

typedef unsigned int       u32;
typedef unsigned long long u64;
typedef __attribute__((ext_vector_type(4))) u32 u32x4;
typedef __attribute__((ext_vector_type(8))) u32 u32x8;

__global__ __launch_bounds__(DIM) void fold27_kernel(const float* __restrict__ x,
                                                     float* __restrict__ out) {
    // LDS rows in kx-major order: r = kx*9 + kz*3 + ky (matches TDM Z-outer/Y-inner)
    __shared__ __align__(16) float rows[NCH][DIM];

    const int t = threadIdx.x;          // output x position, 0..95
    int blk = blockIdx.x;               // (b*96 + z)*96 + y
    const int y = blk % DIM; blk /= DIM;
    const int z = blk % DIM; blk /= DIM;
    const int b = blk;                  // 0..1

    const bool interior = (z >= 1) & (z <= DIM - 2) & (y >= 1) & (y <= DIM - 2);

    if (interior) {
        // ---- TDM path: all 27 source rows in-bounds. Wave 0 issues 3 DMAs. ----
        if ((t >> 5) == 0) {
            constexpr u64 STRIDE_KY = (u64)(3 * CS - DIM);        // dim0 (Y=ky) stride, elems
            constexpr u64 STRIDE_KZ = (u64)(9 * CS - DIM * DIM);  // dim1 (Z=kz) stride, elems
            const u32 lds0 = (u32)(size_t)(&rows[0][0]);
            #pragma unroll
            for (int kx = 0; kx < 3; ++kx) {
                // tile origin: channel c=kx (kz=0,ky=0), iz=z+1, iy=y+1, ix=0
                const u64 e0 = ((u64)(((b * NCH + kx) * DIM + (z + 1)) * DIM + (y + 1))) * DIM;
                const u64 ga = (u64)(size_t)x + e0 * 4u;

                u32x4 g0;
                g0[0] = 1u;                                       // count=1, user descriptor
                g0[1] = lds0 + (u32)(kx * 9 * DIM * 4);           // lds_addr (bytes)
                g0[2] = (u32)(ga & 0xFFFFFFFFu);                  // global_addr[31:0]
                g0[3] = (u32)((ga >> 32) & 0x01FFFFFFu)           // global_addr[56:32]
                      | (2u << 30);                               // type=2 ("image")

                u32x8 g1;
                g1[0] = (2u << 16);                               // data_size=2 -> 4 bytes
                g1[1] = ((u32)DIM << 16);                         // tensor_dim0[15:0]=96
                g1[2] = (3u << 16);                               // t_dim0[31:16]=0 | tensor_dim1[15:0]=3
                g1[3] = ((u32)DIM << 16);                         // t_dim1[31:16]=0 | tile_dim0=96
                g1[4] = 3u | (3u << 16);                          // tile_dim1=3 | tile_dim2=3
                g1[5] = (u32)(STRIDE_KY & 0xFFFFFFFFu);           // dim0_stride[31:0]
                g1[6] = (u32)((STRIDE_KY >> 32) & 0xFFFFu)        // dim0_stride[47:32]
                      | ((u32)(STRIDE_KZ & 0xFFFFu) << 16);       // dim1_stride[15:0]
                g1[7] = (u32)((STRIDE_KZ >> 16) & 0xFFFFFFFFu);   // dim1_stride[47:16]

                u32x4 g2;
                g2[0] = 3u;                                       // tensor_dim2=3
                g2[1] = 0u; g2[2] = 0u; g2[3] = 0u;               // no dim3, no iterate
                u32x4 g3 = {0u, 0u, 0u, 0u};

                asm volatile("tensor_load_to_lds %0, %1, %2, %3"
                             :
                             : "s"(g0), "s"(g1), "s"(g2), "s"(g3)
                             : "memory");
            }
            __builtin_amdgcn_s_wait_tensorcnt(0);
        }
    } else {
        // ---- boundary path: per-lane async copies, zero-fill OOB rows. ----
        const u32 lds_base = (u32)(size_t)(&rows[0][0]);
        #pragma unroll
        for (int r = 0; r < NCH; ++r) {
            const int kx = r / 9, kz = (r / 3) % 3, ky = r % 3;
            const int c  = kz * 9 + ky * 3 + kx;                  // source channel
            const int iz = z + 1 - kz, iy = y + 1 - ky;
            const u32 lds_addr = lds_base + (u32)((r * DIM + t) * 4);
            if ((unsigned)iz < (unsigned)DIM && (unsigned)iy < (unsigned)DIM) {
                const int goff = ((((b * NCH + c) * DIM + iz) * DIM + iy) * DIM + t) * 4;
                asm volatile("global_load_async_to_lds_b32 %0, %1, %2"
                             :
                             : "v"(lds_addr), "v"(goff), "s"(x)
                             : "memory");
            } else {
                rows[r][t] = 0.0f;
            }
        }
        asm volatile("s_wait_asynccnt 0x0" ::: "memory");
    }
    __syncthreads();

    // 27-tap accumulation; x-shift via clamped index + branch-free select.
    float sum = 0.0f;
    #pragma unroll
    for (int r = 0; r < NCH; ++r) {
        const int kx  = r / 9;
        const int idx = t + 1 - kx;                               // in [-1, 96]
        const int idc = idx < 0 ? 0 : (idx > DIM - 1 ? DIM - 1 : idx);
        const float v = rows[r][idc];
        sum += ((unsigned)idx < (unsigned)DIM) ? v : 0.0f;        // folds away for kx=1
    }
    out[((b * DIM + z) * DIM + y) * DIM + t] = sum;
}

extern "C" void kernel_launch(void* const* d_in, const int* in_sizes, int n_in,
                              void* d_out, int out_size, void* d_ws, size_t ws_size,
                              hipStream_t stream) {
    (void)in_sizes; (void)n_in; (void)out_size; (void)d_ws; (void)ws_size;
    const float* x = (const float*)d_in[0];
    float* out = (float*)d_out;
    fold27_kernel<<<dim3(NBLK), dim3(DIM), 0, stream>>>(x, out);
}